// HeteroEncoder_32822140076811
// MI455X (gfx1250) — compile-verified
//
#include <hip/hip_runtime.h>
#include <math.h>

typedef __attribute__((ext_vector_type(2))) float v2f;
typedef __attribute__((ext_vector_type(8))) float v8f;

#define NSN 50000
#define NDN 50000
#define NE  500000
#define HIDF 128
#define OUTF 64
#define NH 8
#define DH 16
#define LSTMH 32
#define NEGS 0.2f

// ---------------------------------------------------------------- LSTM + concat
// xcat[n] = [ x_stream[n][0:16] , lstm_h(x_stream[n][16]) (32) ]  -> 48 feats
__global__ void lstm_concat_kernel(const float* __restrict__ xs,
                                   const float* __restrict__ Wih,
                                   const float* __restrict__ bih,
                                   const float* __restrict__ bhh,
                                   float* __restrict__ xcat) {
    int n = blockIdx.x * blockDim.x + threadIdx.x;
    if (n >= NSN) return;
    const float* row = xs + (size_t)n * 17;
    float xr = row[16];
    float* out = xcat + (size_t)n * 48;
    #pragma unroll
    for (int j = 0; j < 16; ++j) out[j] = row[j];
    for (int j = 0; j < LSTMH; ++j) {
        float gi = xr * Wih[j]      + bih[j]      + bhh[j];        // i gate
        float gg = xr * Wih[64 + j] + bih[64 + j] + bhh[64 + j];   // g gate
        float go = xr * Wih[96 + j] + bih[96 + j] + bhh[96 + j];   // o gate
        float si = 1.f / (1.f + __expf(-gi));
        float so = 1.f / (1.f + __expf(-go));
        float c  = si * tanhf(gg);
        out[16 + j] = so * tanhf(c);
    }
}

// ---------------------------------------------------------------- WMMA fp32 GEMM (weight-stationary)
// Y[N x M] = act( X[N x K] @ W[M x K]^T + bias[M] ),  M = 16*MT
// blockDim = 32*MT (one wave per 16-col tile). B fragments for the whole K dim
// are preloaded into registers once, then reused across a grid-stride loop over
// 16-row tiles. EXEC stays all-ones (uniform control flow) as WMMA requires.
// RELU is a template parameter so the epilogue folds to plain add+max.
template <int K, int MT, int RELU>
__global__ __launch_bounds__(32 * MT)
void wmma_gemm_kernel(const float* __restrict__ X,
                      const float* __restrict__ W,
                      const float* __restrict__ bias,
                      float* __restrict__ Y,
                      int nTiles) {
    constexpr int KF = K / 4;      // WMMA k-steps
    constexpr int M  = MT * 16;
    int lane = threadIdx.x & 31;
    int wave = threadIdx.x >> 5;
    int col0 = wave << 4;
    int mr   = lane & 15;          // A row / B col / bias col within tile
    int koff = (lane >> 4) << 1;   // 0 or 2 (ISA 16x4 f32 fragment layout)

    // preload weight fragments (register-resident for all row tiles)
    v2f bf[KF];
    const float* wp = W + (size_t)(col0 + mr) * K + koff;
    #pragma unroll
    for (int i = 0; i < KF; ++i) bf[i] = *(const v2f*)(wp + 4 * i);

    float bv = bias[col0 + mr];

    for (int tile = blockIdx.x; tile < nTiles; tile += gridDim.x) {
        int row0 = tile << 4;
        const float* xp = X + (size_t)(row0 + mr) * K + koff;
        // prefetch next tile's A fragment (global_prefetch path)
        int nt = tile + gridDim.x;
        if (nt < nTiles)
            __builtin_prefetch(X + (size_t)((nt << 4) + mr) * K + koff, 0, 0);

        v8f c = {0.f, 0.f, 0.f, 0.f, 0.f, 0.f, 0.f, 0.f};
        #pragma unroll
        for (int i = 0; i < KF; ++i) {
            v2f a = *(const v2f*)(xp + 4 * i);
            c = __builtin_amdgcn_wmma_f32_16x16x4_f32(false, a, false, bf[i],
                                                      (short)0, c, false, false);
        }

        int rbase = row0 + ((lane >> 4) << 3);
        float* yp = Y + (size_t)rbase * M + col0 + mr;
        #pragma unroll
        for (int r = 0; r < 8; ++r) {
            float v = c[r] + bv;
            if (RELU) v = fmaxf(v, 0.f);
            yp[(size_t)r * M] = v;
        }
    }
}

// ---------------------------------------------------------------- per-node head scores
// out[n][h] = sum_d x[n][h*16+d] * att[h][d]
__global__ void att_score_kernel(const float* __restrict__ x,
                                 const float* __restrict__ att,
                                 float* __restrict__ out, int N) {
    int t = blockIdx.x * blockDim.x + threadIdx.x;
    if (t >= N * NH) return;
    int n = t >> 3, h = t & 7;
    const float* xp = x + (size_t)n * HIDF + h * DH;
    const float* ap = att + h * DH;
    float s = 0.f;
    #pragma unroll
    for (int d = 0; d < DH; ++d) s += xp[d] * ap[d];
    out[t] = s;
}

// ---------------------------------------------------------------- edge softmax pass 1
// ssum[dst][h] += exp(leaky_relu(asrc[src][h] + adst[dst][h]))
__global__ void edge_sum_kernel(const int* __restrict__ src,
                                const int* __restrict__ dst,
                                const float* __restrict__ asrc,
                                const float* __restrict__ adst,
                                float* __restrict__ ssum) {
    int t = blockIdx.x * blockDim.x + threadIdx.x;
    if (t >= NE * NH) return;
    int e = t >> 3, h = t & 7;
    int s = src[e], d = dst[e];
    float a = asrc[s * NH + h] + adst[d * NH + h];
    a = (a > 0.f) ? a : NEGS * a;
    atomicAdd(&ssum[d * NH + h], __expf(a));
}

// ---------------------------------------------------------------- edge softmax pass 2 + aggregate
// 128 threads per edge (one per feature): agg[dst][f] += x_src[src][f] * alpha[h]
__global__ __launch_bounds__(256)
void edge_agg_kernel(const int* __restrict__ src,
                     const int* __restrict__ dst,
                     const float* __restrict__ asrc,
                     const float* __restrict__ adst,
                     const float* __restrict__ ssum,
                     const float* __restrict__ xsrc,
                     float* __restrict__ agg) {
    long long t = (long long)blockIdx.x * blockDim.x + threadIdx.x;
    int e = (int)(t >> 7);
    int f = (int)(t & 127);
    if (e >= NE) return;
    int h = f >> 4;
    int s = src[e], d = dst[e];
    float a = asrc[s * NH + h] + adst[d * NH + h];
    a = (a > 0.f) ? a : NEGS * a;
    float alpha = __expf(a) / (ssum[d * NH + h] + 1e-16f);
    atomicAdd(&agg[(size_t)d * HIDF + f], xsrc[(size_t)s * HIDF + f] * alpha);
}

// ---------------------------------------------------------------- in-place relu
__global__ void relu_kernel(float* __restrict__ x, int count) {
    int t = blockIdx.x * blockDim.x + threadIdx.x;
    if (t < count) x[t] = fmaxf(x[t], 0.f);
}

// ---------------------------------------------------------------- driver
extern "C" void kernel_launch(void* const* d_in, const int* in_sizes, int n_in,
                              void* d_out, int out_size, void* d_ws, size_t ws_size,
                              hipStream_t stream) {
    (void)in_sizes; (void)n_in; (void)out_size; (void)ws_size;

    const float* x_stream = (const float*)d_in[0];
    const float* x_device = (const float*)d_in[1];
    const int*   edge_sd  = (const int*)d_in[2];   // [0]=stream src, [1]=device dst
    const int*   edge_ds  = (const int*)d_in[3];   // [0]=device src, [1]=stream dst
    const float* lstm_Wih = (const float*)d_in[4];
    const float* lstm_bih = (const float*)d_in[5];
    const float* lstm_bhh = (const float*)d_in[6];
    const float* sc_W     = (const float*)d_in[7];
    const float* sc_b     = (const float*)d_in[8];
    const float* dev_W    = (const float*)d_in[9];
    const float* dev_b    = (const float*)d_in[10];
    const float* proj_W   = (const float*)d_in[11];  // (2,2,128,128)
    const float* proj_b   = (const float*)d_in[12];  // (2,2,128)
    const float* att_src  = (const float*)d_in[13];  // (2,2,8,16)
    const float* att_dst  = (const float*)d_in[14];
    // d_in[15..17] = q, k_W, k_b : semantic attention over a singleton -> identity, elided
    const float* outp_W1  = (const float*)d_in[18];  // (2,64,128)
    const float* outp_b1  = (const float*)d_in[19];
    const float* outp_W2  = (const float*)d_in[20];  // (2,64,64)
    const float* outp_b2  = (const float*)d_in[21];
    float* out = (float*)d_out;

    // workspace layout (floats)
    float* P0  = (float*)d_ws;                       // NSN*128 : hs stream / agg stream
    float* P1  = P0  + (size_t)NSN * HIDF;           // NDN*128 : hs device / agg device
    float* P2  = P1  + (size_t)NDN * HIDF;           // NSN*128 : x0 (projected stream)
    float* P3  = P2  + (size_t)NSN * HIDF;           // NDN*128 : x1 (projected device)
    float* XC  = P3  + (size_t)NDN * HIDF;           // NSN*48
    float* AS0 = XC  + (size_t)NSN * 48;             // NSN*8
    float* AD0 = AS0 + (size_t)NSN * NH;             // NDN*8
    float* AS1 = AD0 + (size_t)NDN * NH;             // NDN*8
    float* AD1 = AS1 + (size_t)NDN * NH;             // NSN*8
    float* SUM = AD1 + (size_t)NSN * NH;             // max(NSN,NDN)*8
    float* H1  = SUM + (size_t)NSN * NH;             // max*64 (MLP hidden)

    const int* sd_src = edge_sd;            const int* sd_dst = edge_sd + NE;
    const int* ds_src = edge_ds;            const int* ds_dst = edge_ds + NE;

    const int NT = NSN / 16;                // 3125 row tiles (N==50000 both types)
    const int GB = 625;                     // grid-stride blocks: 5 tiles per block

    // 1) LSTM + concat
    lstm_concat_kernel<<<(NSN + 255) / 256, 256, 0, stream>>>(
        x_stream, lstm_Wih, lstm_bih, lstm_bhh, XC);

    // 2) encoders: h_s = relu(XC @ sc_W^T + b) ; h_d = relu(x_device @ dev_W^T + b)
    wmma_gemm_kernel<48, 8, 1><<<GB, 256, 0, stream>>>(XC, sc_W, sc_b, P0, NT);
    wmma_gemm_kernel<32, 8, 1><<<GB, 256, 0, stream>>>(x_device, dev_W, dev_b, P1, NT);

    // 3) GAT layers
    for (int l = 0; l < 2; ++l) {
        const float* pw0 = proj_W + (size_t)(l * 2 + 0) * HIDF * HIDF;
        const float* pw1 = proj_W + (size_t)(l * 2 + 1) * HIDF * HIDF;
        const float* pb0 = proj_b + (size_t)(l * 2 + 0) * HIDF;
        const float* pb1 = proj_b + (size_t)(l * 2 + 1) * HIDF;

        wmma_gemm_kernel<128, 8, 0><<<GB, 256, 0, stream>>>(P0, pw0, pb0, P2, NT);
        wmma_gemm_kernel<128, 8, 0><<<GB, 256, 0, stream>>>(P1, pw1, pb1, P3, NT);

        const float* as0 = att_src + (size_t)(l * 2 + 0) * HIDF;  // e=0 (stream->device)
        const float* ad0 = att_dst + (size_t)(l * 2 + 0) * HIDF;
        const float* as1 = att_src + (size_t)(l * 2 + 1) * HIDF;  // e=1 (device->stream)
        const float* ad1 = att_dst + (size_t)(l * 2 + 1) * HIDF;

        att_score_kernel<<<(NSN * NH + 255) / 256, 256, 0, stream>>>(P2, as0, AS0, NSN);
        att_score_kernel<<<(NDN * NH + 255) / 256, 256, 0, stream>>>(P3, ad0, AD0, NDN);
        att_score_kernel<<<(NDN * NH + 255) / 256, 256, 0, stream>>>(P3, as1, AS1, NDN);
        att_score_kernel<<<(NSN * NH + 255) / 256, 256, 0, stream>>>(P2, ad1, AD1, NSN);

        hipMemsetAsync(P0, 0, (size_t)NSN * HIDF * sizeof(float), stream);
        hipMemsetAsync(P1, 0, (size_t)NDN * HIDF * sizeof(float), stream);

        // edge type 0: stream(src, x=P2) -> device(dst), agg into P1
        hipMemsetAsync(SUM, 0, (size_t)NDN * NH * sizeof(float), stream);
        edge_sum_kernel<<<(NE * NH + 255) / 256, 256, 0, stream>>>(sd_src, sd_dst, AS0, AD0, SUM);
        edge_agg_kernel<<<(int)(((long long)NE * HIDF) / 256), 256, 0, stream>>>(
            sd_src, sd_dst, AS0, AD0, SUM, P2, P1);
        relu_kernel<<<(NDN * HIDF + 255) / 256, 256, 0, stream>>>(P1, NDN * HIDF);

        // edge type 1: device(src, x=P3) -> stream(dst), agg into P0
        hipMemsetAsync(SUM, 0, (size_t)NSN * NH * sizeof(float), stream);
        edge_sum_kernel<<<(NE * NH + 255) / 256, 256, 0, stream>>>(ds_src, ds_dst, AS1, AD1, SUM);
        edge_agg_kernel<<<(int)(((long long)NE * HIDF) / 256), 256, 0, stream>>>(
            ds_src, ds_dst, AS1, AD1, SUM, P3, P0);
        relu_kernel<<<(NSN * HIDF + 255) / 256, 256, 0, stream>>>(P0, NSN * HIDF);
    }

    // 4) output MLPs: relu(128->64) then 64->64, per node type
    // type 0 (stream)
    wmma_gemm_kernel<128, 4, 1><<<GB, 128, 0, stream>>>(P0, outp_W1, outp_b1, H1, NT);
    wmma_gemm_kernel<64, 4, 0><<<GB, 128, 0, stream>>>(H1, outp_W2, outp_b2, out, NT);
    // type 1 (device)
    wmma_gemm_kernel<128, 4, 1><<<GB, 128, 0, stream>>>(
        P1, outp_W1 + (size_t)OUTF * HIDF, outp_b1 + OUTF, H1, NT);
    wmma_gemm_kernel<64, 4, 0><<<GB, 128, 0, stream>>>(
        H1, outp_W2 + (size_t)OUTF * OUTF, outp_b2 + OUTF,
        out + (size_t)NSN * OUTF, NT);
}